// ManualSparseQuantLinear_5257039970427
// MI455X (gfx1250) — compile-verified
//
#include <hip/hip_runtime.h>

// ---------------------------------------------------------------------------
// MI455X (gfx1250): SmoothQuant + NVFP4 fake-quant + 2:4-sparse SWMMAC GEMM
// + LoRA + bias, fully fused.
//   out[M,N] = Xq[M,K] * W[N,K]^T + (Xs*LA^T)*LB^T + bias
// W is 2:4 magnitude-sparse along K -> main loop uses
// v_swmmac_f32_16x16x64_f16 (A = compressed W + indices, B = dense Xq).
// LoRA folded in by extending K: expanded 4096->4160, compressed 2048->2080.
// Async global->LDS staging (ASYNCcnt), double-buffered.
// ---------------------------------------------------------------------------

typedef __attribute__((ext_vector_type(8)))  _Float16 v8h;
typedef __attribute__((ext_vector_type(16))) _Float16 v16h;
typedef __attribute__((ext_vector_type(32))) _Float16 v32h;
typedef __attribute__((ext_vector_type(8)))  float    v8f;
typedef int v4i __attribute__((vector_size(4 * sizeof(int))));

#define M_DIM 4096   // B*S
#define K_DIM 4096   // IN_F
#define N_DIM 4096   // OUT_F
#define RANK  32
#define KE    4160   // expanded K incl. LoRA block (65 steps of 64)
#define KC    2080   // compressed K (KE/2)
#define KI    130    // index dwords per row (KE/32)
#define NSTEP 65     // K iterations

#define NBLK 128     // block tile, A-side (N / sparse W rows)
#define MBLK 256     // block tile, B-side (M / Xq rows)
#define ASTRIDE 40   // LDS halfs/row for compressed A (32 data + 8 pad)
#define BSTRIDE 72   // LDS halfs/row for B (64 data + 8 pad; 144B bank-conflict-free)
#define LDS_A (NBLK * ASTRIDE)    // 5120 halfs
#define LDS_B (MBLK * BSTRIDE)    // 18432 halfs
#define STAGE (LDS_A + LDS_B)     // 23552 halfs per pipeline stage (~46KB)

#define GLOBAL_AS __attribute__((address_space(1)))
#define LDS_AS    __attribute__((address_space(3)))

#if __has_builtin(__builtin_amdgcn_global_load_async_to_lds_b128)
#define HAVE_ASYNC_LDS 1
#define ASYNC_CP16(g, l, OFF) \
  __builtin_amdgcn_global_load_async_to_lds_b128((GLOBAL_AS v4i*)(g), (LDS_AS v4i*)(l), (OFF), 0)
#endif

#if __has_builtin(__builtin_amdgcn_s_wait_asynccnt)
#define WAIT_ASYNC() __builtin_amdgcn_s_wait_asynccnt(0)
#else
#define WAIT_ASYNC() asm volatile("s_wait_asynccnt 0" ::: "memory")
#endif

static __device__ __forceinline__ v16h cat16(v8h lo, v8h hi) {
  return __builtin_shufflevector(lo, hi, 0,1,2,3,4,5,6,7,8,9,10,11,12,13,14,15);
}
static __device__ __forceinline__ v32h cat32(v16h lo, v16h hi) {
  return __builtin_shufflevector(lo, hi, 0,1,2,3,4,5,6,7,8,9,10,11,12,13,14,15,
                                 16,17,18,19,20,21,22,23,24,25,26,27,28,29,30,31);
}

// ---------------------------------------------------------------------------
// Kernel 1: per-row SmoothQuant + NVFP4 fake-quant (f16) + LoRA stage-1.
// ---------------------------------------------------------------------------
__global__ __launch_bounds__(256)
void prep_x_kernel(const float* __restrict__ x, const float* __restrict__ ss,
                   const float* __restrict__ la, _Float16* __restrict__ xq,
                   float* __restrict__ tbuf) {
  const int row  = blockIdx.x;
  const int t    = threadIdx.x;
  const int col0 = t * 16;

  __shared__ float lt[RANK];
  if (t < RANK) lt[t] = 0.0f;
  __syncthreads();

  const float4* xp = (const float4*)(x + (size_t)row * K_DIM + col0);
  const float4* sp = (const float4*)(ss + col0);

  float xs[16];
  #pragma unroll
  for (int q = 0; q < 4; ++q) {
    float4 xv = xp[q]; float4 sv = sp[q];
    xs[q*4+0] = xv.x * sv.x; xs[q*4+1] = xv.y * sv.y;
    xs[q*4+2] = xv.z * sv.z; xs[q*4+3] = xv.w * sv.w;
  }

  float amax = 1e-12f;
  #pragma unroll
  for (int j = 0; j < 16; ++j) amax = fmaxf(amax, fabsf(xs[j]));
  const float scale = amax * (1.0f / 6.0f);
  const float inv   = 6.0f / amax;

  _Float16 q16[16];
  #pragma unroll
  for (int j = 0; j < 16; ++j) {
    const float xn = fabsf(xs[j]) * inv;             // in [0, 6]
    const float lv = (xn <= 0.25f) ? 0.0f :          // ties -> lower level
                     (xn <= 0.75f) ? 0.5f :
                     (xn <= 1.25f) ? 1.0f :
                     (xn <= 1.75f) ? 1.5f :
                     (xn <= 2.5f ) ? 2.0f :
                     (xn <= 3.5f ) ? 3.0f :
                     (xn <= 5.0f ) ? 4.0f : 6.0f;
    q16[j] = (_Float16)copysignf(lv * scale, xs[j]);
  }
  v8h* outp = (v8h*)(xq + (size_t)row * KE + col0);
  outp[0] = *(v8h*)&q16[0];
  outp[1] = *(v8h*)&q16[8];

  #pragma unroll 4
  for (int r = 0; r < RANK; ++r) {
    const float4* lap = (const float4*)(la + (size_t)r * K_DIM + col0);
    float s = 0.0f;
    #pragma unroll
    for (int q = 0; q < 4; ++q) {
      float4 lv = lap[q];
      s += xs[q*4+0]*lv.x + xs[q*4+1]*lv.y + xs[q*4+2]*lv.z + xs[q*4+3]*lv.w;
    }
    #pragma unroll
    for (int off = 16; off > 0; off >>= 1) s += __shfl_xor(s, off, 32);
    if ((t & 31) == 0) atomicAdd(&lt[r], s);
  }
  __syncthreads();
  if (t < RANK) tbuf[(size_t)row * RANK + t] = lt[t];
}

// ---------------------------------------------------------------------------
// Kernel 2: 2:4 compression of the fake-quantized sparse weight.
// Thread t handles 4 groups (16 expanded cols): emits 8 f16 values + 16 idx bits.
// Index layout (ISA 7.12.4): 4 bits/group = idx0 | idx1<<2, idx0 < idx1.
// ---------------------------------------------------------------------------
__global__ __launch_bounds__(256)
void prep_w_sparse_kernel(const float* __restrict__ wsrc, _Float16* __restrict__ wc,
                          unsigned short* __restrict__ iw16) {
  const int n = blockIdx.x;
  const int t = threadIdx.x;
  const int col0 = t * 16;
  const float4* sp = (const float4*)(wsrc + (size_t)n * K_DIM + col0);

  _Float16 cvals[8];
  unsigned int codes = 0;
  #pragma unroll
  for (int g = 0; g < 4; ++g) {
    float4 v = sp[g];
    float vv[4] = {v.x, v.y, v.z, v.w};
    int i0 = -1, i1 = -1;
    #pragma unroll
    for (int j = 0; j < 4; ++j) {
      if (vv[j] != 0.0f) { if (i0 < 0) i0 = j; else if (i1 < 0) i1 = j; }
    }
    if (i0 < 0)      { i0 = 0; i1 = 1; }
    else if (i1 < 0) { if (i0 == 3) { i0 = 2; i1 = 3; } else { i1 = i0 + 1; } }
    cvals[2*g]   = (_Float16)vv[i0];
    cvals[2*g+1] = (_Float16)vv[i1];
    codes |= (unsigned)(i0 | (i1 << 2)) << (4 * g);
  }
  *(v8h*)(wc + (size_t)n * KC + t * 8) = *(v8h*)cvals;
  iw16[(size_t)n * (KI * 2) + t] = (unsigned short)codes;
}

// ---------------------------------------------------------------------------
// Kernel 3: LoRA fold-in.
//  B-side: Xq[:,4096+4g+j] = T[:,2g+j] (j<2) else 0  (2:4 spread)
//  A-side: Wc[:,2048+r] = LB[:,r]; index dwords = 0x44444444 (idx pairs (0,1))
// ---------------------------------------------------------------------------
__global__ __launch_bounds__(256)
void append_kernel(const float* __restrict__ tbuf, const float* __restrict__ lb,
                   _Float16* __restrict__ xq, _Float16* __restrict__ wc,
                   unsigned int* __restrict__ iw) {
  const int i = blockIdx.x * 256 + threadIdx.x;     // 0 .. M*64-1
  const int m = i >> 6, col = i & 63;
  const int g = col >> 2, j = col & 3;
  const float v = (j < 2) ? tbuf[m * RANK + 2 * g + j] : 0.0f;
  xq[(size_t)m * KE + K_DIM + col] = (_Float16)v;
  if (i < N_DIM * RANK) {
    const int n = i >> 5, r = i & 31;
    wc[(size_t)n * KC + (K_DIM / 2) + r] = (_Float16)lb[n * RANK + r];
  }
  if (i < N_DIM * 2) {
    const int n = i >> 1, d = i & 1;
    iw[(size_t)n * KI + (K_DIM / 32) + d] = 0x44444444u;
  }
}

// ---------------------------------------------------------------------------
// Kernel 4: 2:4-sparse SWMMAC GEMM.  out[m,n] = sum_k Xq[m,k]*W[n,k] + bias[n]
// D = A(sparse W) x B(dense Xq): D tile is [n][m] -> contiguous-n epilogue.
// 256 threads = 8 waves (2 A-side x 4 B-side), block 128(N) x 256(M),
// wave tile 64x64 (4x4 tiles), expanded-K step 64, async double-buffered.
// ---------------------------------------------------------------------------
__global__ __launch_bounds__(256)
void gemm_swmmac_kernel(const _Float16* __restrict__ Xq, const _Float16* __restrict__ Wc,
                        const unsigned int* __restrict__ Iw, const float* __restrict__ bias,
                        float* __restrict__ out) {
  __shared__ _Float16 ls[2 * STAGE];   // ~92 KB

  const int tid  = threadIdx.x;
  const int lane = tid & 31;
  const int wave = tid >> 5;
  const int waveA = wave >> 2;         // 0..1 : 64-row slab of N
  const int waveB = wave & 3;          // 0..3 : 64-col slab of M
  const int nblk = blockIdx.y * NBLK;
  const int mblk = blockIdx.x * MBLK;

  v8f c[4][4];                          // [nt][mt]
  #pragma unroll
  for (int i = 0; i < 4; ++i)
    #pragma unroll
    for (int j = 0; j < 4; ++j)
      #pragma unroll
      for (int v = 0; v < 8; ++v) c[i][j][v] = 0.0f;

  // staging maps
  const int arow = tid >> 1;
  const int aseg = (tid & 1) * 16;                    // halfs
  const _Float16* gA = Wc + (size_t)(nblk + arow) * KC + aseg;
  const _Float16* gB = Xq + (size_t)(mblk + tid) * KE;
  _Float16* sA0 = &ls[arow * ASTRIDE + aseg];
  _Float16* sB0 = &ls[LDS_A + tid * BSTRIDE];

  // per-lane fragment maps
  const int lh   = lane & 15;
  const int hsel = lane >> 4;                          // 0/1
  const _Float16* rA0 = &ls[(waveA * 64 + lh) * ASTRIDE + hsel * 8];
  const _Float16* rB0 = &ls[LDS_A + (waveB * 64 + lh) * BSTRIDE + hsel * 16];
  const unsigned int* gI = Iw + (size_t)(nblk + waveA * 64 + lh) * KI + hsel;

  auto stage = [&](int pb, int s) {
    const _Float16* ga = gA + s * 32;                  // compressed step
    const _Float16* gb = gB + s * 64;                  // expanded step
    _Float16* sa = sA0 + pb * STAGE;
    _Float16* sb = sB0 + pb * STAGE;
#ifdef HAVE_ASYNC_LDS
    ASYNC_CP16(ga, sa, 0);
    ASYNC_CP16(ga, sa, 16);
    ASYNC_CP16(gb, sb, 0);
    ASYNC_CP16(gb, sb, 16);
    ASYNC_CP16(gb, sb, 32);
    ASYNC_CP16(gb, sb, 48);
    ASYNC_CP16(gb, sb, 64);
    ASYNC_CP16(gb, sb, 80);
    ASYNC_CP16(gb, sb, 96);
    ASYNC_CP16(gb, sb, 112);
#else
    v8h a0 = *(const v8h*)(ga), a1 = *(const v8h*)(ga + 8);
    *(v8h*)(sa) = a0; *(v8h*)(sa + 8) = a1;
    #pragma unroll
    for (int q = 0; q < 8; ++q) {
      v8h b = *(const v8h*)(gb + q * 8);
      *(v8h*)(sb + q * 8) = b;
    }
#endif
  };

  stage(0, 0);
  int p = 0;
  for (int s = 0; s < NSTEP; ++s) {
    WAIT_ASYNC();          // in-order ASYNCcnt: buffer p complete
    __syncthreads();       // all waves past reads of buffer p^1

    if (s + 1 < NSTEP) {
      stage(p ^ 1, s + 1);
      __builtin_prefetch((const void*)(gA + s * 32 + 64), 0, 3);
      __builtin_prefetch((const void*)(gB + s * 64 + 128), 0, 3);
    }

    const _Float16* ra = rA0 + p * STAGE;
    const _Float16* rb = rB0 + p * STAGE;

    // preload 4 compressed-A fragments + index dwords (A reused across all mt)
    v16h af[4];
    int  ix[4];
    #pragma unroll
    for (int nt = 0; nt < 4; ++nt) {
      const _Float16* qa = ra + nt * 16 * ASTRIDE;
      af[nt] = cat16(*(const v8h*)qa, *(const v8h*)(qa + 16));
      ix[nt] = (int)gI[(size_t)(nt * 16) * KI + s * 2];
    }
    #pragma unroll
    for (int mt = 0; mt < 4; ++mt) {
      // dense B 64x16: lane -> col m=lh, K-quarter by hsel: chunks at +0B and +64B
      const _Float16* qb = rb + mt * 16 * BSTRIDE;
      v16h blo = cat16(*(const v8h*)qb,        *(const v8h*)(qb + 8));
      v16h bhi = cat16(*(const v8h*)(qb + 32), *(const v8h*)(qb + 40));
      v32h bf  = cat32(blo, bhi);
      #pragma unroll
      for (int nt = 0; nt < 4; ++nt) {
        // (neg_a, A_compressed, neg_b, B_dense, C, index, reuse_a, reuse_b)
        c[nt][mt] = __builtin_amdgcn_swmmac_f32_16x16x64_f16(
            false, af[nt], false, bf, c[nt][mt], ix[nt], false, false);
      }
    }
    p ^= 1;
  }

  // epilogue: D[n][m]; lane -> m = lh, n = nbase + hsel*8 + v (contiguous v8f)
  #pragma unroll
  for (int nt = 0; nt < 4; ++nt) {
    const int nbase = nblk + waveA * 64 + nt * 16 + hsel * 8;
    const v8f bv = *(const v8f*)(bias + nbase);
    #pragma unroll
    for (int mt = 0; mt < 4; ++mt) {
      const int m = mblk + waveB * 64 + mt * 16 + lh;
      v8f r = c[nt][mt] + bv;
      *(v8f*)(out + (size_t)m * N_DIM + nbase) = r;
    }
  }
}

// ---------------------------------------------------------------------------
extern "C" void kernel_launch(void* const* d_in, const int* in_sizes, int n_in,
                              void* d_out, int out_size, void* d_ws, size_t ws_size,
                              hipStream_t stream) {
  const float* x    = (const float*)d_in[0];   // [2,2048,4096]
  const float* ss   = (const float*)d_in[1];   // [4096]
  const float* wsrc = (const float*)d_in[2];   // [4096,4096] 2:4-sparse fake-quant
  const float* la   = (const float*)d_in[3];   // [32,4096]
  const float* lb   = (const float*)d_in[4];   // [4096,32]
  const float* bias = (const float*)d_in[5];   // [4096]
  float* out = (float*)d_out;

  char* ws = (char*)d_ws;
  const size_t xq_bytes = (size_t)M_DIM * KE * sizeof(_Float16);       // 34,078,720
  const size_t wc_bytes = (size_t)N_DIM * KC * sizeof(_Float16);       // 17,039,360
  const size_t iw_bytes = (size_t)N_DIM * KI * sizeof(unsigned int);   //  2,129,920
  _Float16*     xq   = (_Float16*)ws;
  _Float16*     wc   = (_Float16*)(ws + xq_bytes);
  unsigned int* iw   = (unsigned int*)(ws + xq_bytes + wc_bytes);
  float*        tbuf = (float*)(ws + xq_bytes + wc_bytes + iw_bytes);  // M*RANK f32

  prep_x_kernel<<<M_DIM, 256, 0, stream>>>(x, ss, la, xq, tbuf);
  prep_w_sparse_kernel<<<N_DIM, 256, 0, stream>>>(wsrc, wc, (unsigned short*)iw);
  append_kernel<<<(M_DIM * 64) / 256, 256, 0, stream>>>(tbuf, lb, xq, wc, iw);
  gemm_swmmac_kernel<<<dim3(M_DIM / MBLK, N_DIM / NBLK), 256, 0, stream>>>(xq, wc, iw, bias, out);
}